// Classifier_41970420417053
// MI455X (gfx1250) — compile-verified
//
#include <hip/hip_runtime.h>
#include <math.h>

#define N_NODES 50000
#define N_EDGES 800000
#define G_GRAPHS 128
#define LATD 128
#define HIDD 256
#define NCLS 10

typedef __attribute__((ext_vector_type(2))) float v2f;
typedef __attribute__((ext_vector_type(8))) float v8f;

// ---------------------------------------------------------------------------
// zero fill (float4 granular; counts are multiples of 4)
// ---------------------------------------------------------------------------
__global__ void zero_kernel(float* __restrict__ p, int count4) {
  int t = blockIdx.x * blockDim.x + threadIdx.x;
  if (t < count4) {
    float4 z; z.x = 0.f; z.y = 0.f; z.z = 0.f; z.w = 0.f;
    ((float4*)p)[t] = z;
  }
}

// ---------------------------------------------------------------------------
// out[rows, ncols] = act(A[rows,128] @ W[ncols,128]^T + bias (+ skip))
// fp32 WMMA 16x16x4. Block = 256 threads (8 waves): 16 rows x 128 cols.
// grid.x = rows/16, grid.y = ncols/128.  rows, ncols tile-aligned -> EXEC all 1s.
// out2 (optional, uniform) additionally stores relu(v) (used to produce both
// input_message and cur = relu(input_message) in one pass).
// ---------------------------------------------------------------------------
template <bool RELU, bool SKIP>
__global__ __launch_bounds__(256) void gemm128_wmma(
    const float* __restrict__ A, const float* __restrict__ W,
    const float* __restrict__ bias, const float* __restrict__ skip,
    float* __restrict__ out, float* __restrict__ out2, int ncols) {
  __shared__ float As[16][132];  // stride 132 -> bank-conflict-free fragment reads

  const int row0 = blockIdx.x * 16;
  const int colchunk = blockIdx.y * 128;
  const int tid = threadIdx.x;

  // Stage A tile: 16 rows x 128 K-floats, coalesced float4 loads.
  {
    const float4* Ag = (const float4*)(A + (size_t)row0 * 128);
#pragma unroll
    for (int i = 0; i < 2; ++i) {
      int idx = tid + i * 256;      // 0..511 over 16x32 float4s
      int r = idx >> 5;
      int c4 = idx & 31;
      float4 v = Ag[r * 32 + c4];
      As[r][c4 * 4 + 0] = v.x;
      As[r][c4 * 4 + 1] = v.y;
      As[r][c4 * 4 + 2] = v.z;
      As[r][c4 * 4 + 3] = v.w;
    }
  }
  __syncthreads();

  const int wave = tid >> 5;
  const int lane = tid & 31;
  const int half = lane >> 4;   // 0: lanes 0-15, 1: lanes 16-31
  const int nloc = lane & 15;   // M index for A frag, N index for B frag
  const int kb = half * 2;      // lanes 16-31 hold K+2,K+3
  const int gcol = colchunk + wave * 16 + nloc;

  const float* wrow = W + (size_t)gcol * 128 + kb;  // B[k][n] = W[n][k]
  const float* arow = &As[nloc][kb];

  v8f acc = {};
#pragma unroll
  for (int k0 = 0; k0 < 128; k0 += 4) {
    v2f a = *(const v2f*)(arow + k0);
    v2f b = *(const v2f*)(wrow + k0);
    acc = __builtin_amdgcn_wmma_f32_16x16x4_f32(false, a, false, b, (short)0,
                                                acc, false, false);
  }

  // Epilogue: C/D layout -> VGPR i holds row (i + 8*half), col = lane&15.
  const float bv = bias[gcol];
#pragma unroll
  for (int i = 0; i < 8; ++i) {
    int grow = row0 + half * 8 + i;
    size_t o = (size_t)grow * ncols + gcol;
    float v = acc[i] + bv;
    if (SKIP) v += skip[o];
    out[o] = RELU ? fmaxf(v, 0.0f) : v;
    if (out2) out2[o] = fmaxf(v, 0.0f);
  }
}

// ---------------------------------------------------------------------------
// agg[dst] += cur[src]  (binary-adjacency SpMM). 32 lanes/edge, float4 each.
// ---------------------------------------------------------------------------
__global__ void scatter_add_kernel(const float* __restrict__ cur,
                                   const int* __restrict__ src,
                                   const int* __restrict__ dst,
                                   float* __restrict__ agg, int nedges) {
  int t = blockIdx.x * blockDim.x + threadIdx.x;
  int e = t >> 5;
  if (e >= nedges) return;
  int f = (t & 31) << 2;
  int s = src[e], d = dst[e];
  float4 v = *(const float4*)(cur + (size_t)s * LATD + f);
  float* p = agg + (size_t)d * LATD + f;
  atomicAdd(p + 0, v.x);
  atomicAdd(p + 1, v.y);
  atomicAdd(p + 2, v.z);
  atomicAdd(p + 3, v.w);
}

// ---------------------------------------------------------------------------
// embed[graph_id[n]] += x[n]   (per-graph sum pooling)
// ---------------------------------------------------------------------------
__global__ void pool_add_kernel(const float* __restrict__ x,
                                const int* __restrict__ gid,
                                float* __restrict__ embed, int nnodes) {
  int t = blockIdx.x * blockDim.x + threadIdx.x;
  int node = t >> 5;
  if (node >= nnodes) return;
  int f = (t & 31) << 2;
  int g = gid[node];
  float4 v = *(const float4*)(x + (size_t)node * LATD + f);
  float* p = embed + (size_t)g * LATD + f;
  atomicAdd(p + 0, v.x);
  atomicAdd(p + 1, v.y);
  atomicAdd(p + 2, v.z);
  atomicAdd(p + 3, v.w);
}

__global__ void relu_kernel(float* __restrict__ p, int count) {
  int t = blockIdx.x * blockDim.x + threadIdx.x;
  if (t < count) p[t] = fmaxf(p[t], 0.0f);
}

// ---------------------------------------------------------------------------
// logits = log_softmax(h1 @ w_h2^T + b_h2); nll, loss, acc.
// One block, 128 threads (one per graph). Output layout:
//   out[0..1279]  logits (G x NC)
//   out[1280]     loss
//   out[1281]     acc
// ---------------------------------------------------------------------------
__global__ __launch_bounds__(128) void classifier_kernel(
    const float* __restrict__ h1, const float* __restrict__ w_h2,
    const float* __restrict__ b_h2, const int* __restrict__ labels,
    float* __restrict__ out) {
  __shared__ float w2s[NCLS * HIDD];
  __shared__ float b2s[NCLS];
  __shared__ float red[G_GRAPHS];

  int t = threadIdx.x;  // 0..127 == graph id
  for (int i = t; i < NCLS * HIDD; i += 128) w2s[i] = w_h2[i];
  if (t < NCLS) b2s[t] = b_h2[t];
  __syncthreads();

  const float* hrow = h1 + (size_t)t * HIDD;
  float logit[NCLS];
  float mx = -3.402823466e38f;
#pragma unroll
  for (int c = 0; c < NCLS; ++c) {
    float s = b2s[c];
    const float* wr = w2s + c * HIDD;
    for (int k = 0; k < HIDD; ++k) s += hrow[k] * wr[k];
    logit[c] = s;
    mx = fmaxf(mx, s);
  }
  float sum = 0.0f;
#pragma unroll
  for (int c = 0; c < NCLS; ++c) sum += expf(logit[c] - mx);
  float lse = mx + logf(sum);

  int amax = 0;
  float best = logit[0];
#pragma unroll
  for (int c = 1; c < NCLS; ++c) {
    if (logit[c] > best) { best = logit[c]; amax = c; }
  }
  int lab = labels[t];
#pragma unroll
  for (int c = 0; c < NCLS; ++c) out[t * NCLS + c] = logit[c] - lse;
  float nll = -(logit[lab] - lse);

  red[t] = nll;
  __syncthreads();
  if (t == 0) {
    float s = 0.0f;
    for (int i = 0; i < G_GRAPHS; ++i) s += red[i];
    out[G_GRAPHS * NCLS + 0] = s / (float)G_GRAPHS;
  }
  __syncthreads();
  red[t] = (amax == lab) ? 1.0f : 0.0f;
  __syncthreads();
  if (t == 0) {
    float s = 0.0f;
    for (int i = 0; i < G_GRAPHS; ++i) s += red[i];
    out[G_GRAPHS * NCLS + 1] = s / (float)G_GRAPHS;
  }
}

// ---------------------------------------------------------------------------
extern "C" void kernel_launch(void* const* d_in, const int* in_sizes, int n_in,
                              void* d_out, int out_size, void* d_ws,
                              size_t ws_size, hipStream_t stream) {
  const float* node_feat = (const float*)d_in[0];
  const int*   edge_src  = (const int*)d_in[1];
  const int*   edge_dst  = (const int*)d_in[2];
  const int*   graph_id  = (const int*)d_in[3];
  const int*   labels    = (const int*)d_in[4];
  const float* w_n2l     = (const float*)d_in[5];
  const float* b_n2l     = (const float*)d_in[6];
  const float* w_conv    = (const float*)d_in[7];
  const float* b_conv    = (const float*)d_in[8];
  const float* w_out     = (const float*)d_in[9];
  const float* b_out     = (const float*)d_in[10];
  const float* w_h1      = (const float*)d_in[11];
  const float* b_h1      = (const float*)d_in[12];
  const float* w_h2      = (const float*)d_in[13];
  const float* b_h2      = (const float*)d_in[14];
  float* out = (float*)d_out;

  float* ws    = (float*)d_ws;
  float* IM    = ws;                                   // [N,128] pre-relu input_message
  float* cur   = IM + (size_t)N_NODES * LATD;          // [N,128]
  float* agg   = cur + (size_t)N_NODES * LATD;         // [N,128] (also reused for node out)
  float* embed = agg + (size_t)N_NODES * LATD;         // [G,128]
  float* h1    = embed + (size_t)G_GRAPHS * LATD;      // [G,256]

  dim3 blk(256);
  dim3 gN(N_NODES / 16, 1);

  // input_message = node_feat @ w_n2l^T + b ; cur = relu(input_message)
  gemm128_wmma<false, false><<<gN, blk, 0, stream>>>(node_feat, w_n2l, b_n2l,
                                                     nullptr, IM, cur, LATD);

  const int nz_node = (N_NODES * LATD / 4 + 255) / 256;
  const int ns_edge = (N_EDGES * 32 + 255) / 256;
  for (int lv = 0; lv < 3; ++lv) {
    zero_kernel<<<nz_node, blk, 0, stream>>>(agg, N_NODES * LATD / 4);
    scatter_add_kernel<<<ns_edge, blk, 0, stream>>>(cur, edge_src, edge_dst,
                                                    agg, N_EDGES);
    // cur = relu(agg @ w_conv^T + b_conv + input_message)
    gemm128_wmma<true, true><<<gN, blk, 0, stream>>>(agg, w_conv, b_conv, IM,
                                                     cur, nullptr, LATD);
  }

  // node out = relu(cur @ w_out^T + b_out)  -> stored in agg
  gemm128_wmma<true, false><<<gN, blk, 0, stream>>>(cur, w_out, b_out, nullptr,
                                                    agg, nullptr, LATD);

  // embed = relu(segment_sum(out, graph_id))
  zero_kernel<<<(G_GRAPHS * LATD / 4 + 255) / 256, blk, 0, stream>>>(
      embed, G_GRAPHS * LATD / 4);
  pool_add_kernel<<<(N_NODES * 32 + 255) / 256, blk, 0, stream>>>(
      agg, graph_id, embed, N_NODES);
  relu_kernel<<<(G_GRAPHS * LATD + 255) / 256, blk, 0, stream>>>(
      embed, G_GRAPHS * LATD);

  // h1 = relu(embed @ w_h1^T + b_h1)   [128 x 256]
  gemm128_wmma<true, false><<<dim3(G_GRAPHS / 16, HIDD / 128), blk, 0, stream>>>(
      embed, w_h1, b_h1, nullptr, h1, nullptr, HIDD);

  // logits / loss / acc
  classifier_kernel<<<1, 128, 0, stream>>>(h1, w_h2, b_h2, labels, out);
}